// RadLIFLayer_14663018349101
// MI455X (gfx1250) — compile-verified
//
#include <hip/hip_runtime.h>
#include <hip/hip_bf16.h>
#include <stdint.h>

typedef __attribute__((ext_vector_type(2))) float v2f;
typedef __attribute__((ext_vector_type(8))) float v8f;

#define BB   256
#define TT   100
#define INP  700
#define HH   1024
#define M1   (BB * TT)   // 25600
#define NR   40          // BN partial row-chunks (25600/40 = 640 rows each)
#define BN_EPS 1e-5f
#define THRESH 1.0f

// clip limits (double-evaluated constants, cast to f32 like the reference)
#define ALPHA_LO 0.8187307530779818f   // exp(-1/5)
#define ALPHA_HI 0.9607894391523232f   // exp(-1/25)
#define BETA_LO  0.9672161004820059f   // exp(-1/30)
#define BETA_HI  0.9917012924341500f   // exp(-1/120)

static __device__ __forceinline__ v8f wmma_f32_4(v2f a, v2f b, v8f c) {
    return __builtin_amdgcn_wmma_f32_16x16x4_f32(false, a, false, b, (short)0, c,
                                                 false, false);
}

// ---------------------------------------------------------------------------
// Prep: zero-diag V, copy mutable state, clip neuron parameters.
// ---------------------------------------------------------------------------
__global__ __launch_bounds__(256) void prep_kernel(
    const float* __restrict__ Vw,
    const float* __restrict__ alpha, const float* __restrict__ beta,
    const float* __restrict__ aa, const float* __restrict__ bb,
    const float* __restrict__ ut0, const float* __restrict__ wt0,
    float* __restrict__ Vz, float* __restrict__ ut, float* __restrict__ wt,
    float* __restrict__ alpha_c, float* __restrict__ beta_c,
    float* __restrict__ a_c, float* __restrict__ b_c)
{
    const int idx = blockIdx.x * 256 + threadIdx.x;
    if (idx < HH * HH) {
        const int row = idx >> 10, col = idx & (HH - 1);
        Vz[idx] = (row == col) ? 0.0f : Vw[idx];
    }
    if (idx < BB * HH) {
        ut[idx] = ut0[idx];
        wt[idx] = wt0[idx];
    }
    if (idx < HH) {
        alpha_c[idx] = fminf(fmaxf(alpha[idx], ALPHA_LO), ALPHA_HI);
        beta_c[idx]  = fminf(fmaxf(beta[idx],  BETA_LO),  BETA_HI);
        a_c[idx]     = fminf(fmaxf(aa[idx], -1.0f), 1.0f);
        b_c[idx]     = fminf(fmaxf(bb[idx],  0.0f), 2.0f);
    }
}

// ---------------------------------------------------------------------------
// GEMM1: Wx[m][n] = sum_k x[m][k] * W[n][k]   (M=25600, N=1024, K=700)
// Block 256 thr = 8 waves; block tile 128x128; wave tile 32x64 (2x4 WMMA tiles)
// f32 WMMA 16x16x4; per lane each A/B frag is a contiguous float2 (K-pair).
// Whole working set (x + W + Wx) is L2-resident (192 MB) -> global-direct.
// ---------------------------------------------------------------------------
__global__ __launch_bounds__(256) void gemm_wx_kernel(
    const float* __restrict__ x, const float* __restrict__ W,
    float* __restrict__ Wx)
{
    const int lane   = threadIdx.x & 31;
    const int wid    = threadIdx.x >> 5;
    const int lanelo = lane & 15;
    const int lanehi = lane >> 4;        // K-pair selector
    const int m0 = blockIdx.x * 128 + (wid >> 1) * 32;
    const int n0 = blockIdx.y * 128 + (wid & 1) * 64;

    v8f acc[2][4];
    v8f zero = {0.f, 0.f, 0.f, 0.f, 0.f, 0.f, 0.f, 0.f};
#pragma unroll
    for (int i = 0; i < 2; ++i)
#pragma unroll
        for (int j = 0; j < 4; ++j) acc[i][j] = zero;

    // A frag (16x4 f32): M = lane%16, K = 2*(lane/16) + vgpr
    const float* arow0 = x + (size_t)(m0 + lanelo) * INP + 2 * lanehi;
    const float* arow1 = arow0 + (size_t)16 * INP;
    // B frag: N = lane%16, K = 2*(lane/16)+vgpr; B[k][n] = W[n][k] -> contiguous pair
    const float* brow[4];
#pragma unroll
    for (int j = 0; j < 4; ++j)
        brow[j] = W + (size_t)(n0 + j * 16 + lanelo) * INP + 2 * lanehi;

    for (int k = 0; k < INP; k += 4) {
        v2f a0 = *(const v2f*)(arow0 + k);
        v2f a1 = *(const v2f*)(arow1 + k);
#pragma unroll
        for (int j = 0; j < 4; ++j) {
            v2f b = *(const v2f*)(brow[j] + k);
            acc[0][j] = wmma_f32_4(a0, b, acc[0][j]);
            acc[1][j] = wmma_f32_4(a1, b, acc[1][j]);
        }
    }

    // C layout: lane L, vgpr v -> M = 8*(L/16)+v, N = L%16
#pragma unroll
    for (int i = 0; i < 2; ++i) {
        const int mBase = m0 + i * 16 + 8 * lanehi;
#pragma unroll
        for (int j = 0; j < 4; ++j) {
            const int n = n0 + j * 16 + lanelo;
#pragma unroll
            for (int v = 0; v < 8; ++v)
                Wx[(size_t)(mBase + v) * HH + n] = acc[i][j][v];
        }
    }
}

// ---------------------------------------------------------------------------
// BN reduction, stage 1: deterministic per-row-chunk partial sum / sumsq.
// ---------------------------------------------------------------------------
__global__ __launch_bounds__(256) void bn_partial_kernel(
    const float* __restrict__ Wx, float* __restrict__ psum,
    float* __restrict__ psq)
{
    const int h  = blockIdx.x * 256 + threadIdx.x;
    const int r0 = blockIdx.y * (M1 / NR);
    float s = 0.f, q = 0.f;
    for (int r = r0; r < r0 + M1 / NR; ++r) {
        float v = Wx[(size_t)r * HH + h];
        s += v;
        q += v * v;
    }
    psum[(size_t)blockIdx.y * HH + h] = s;
    psq [(size_t)blockIdx.y * HH + h] = q;
}

// Stage 2: finalize mean/var, fold gamma/beta into per-channel scale+shift.
__global__ __launch_bounds__(256) void bn_final_kernel(
    const float* __restrict__ psum, const float* __restrict__ psq,
    const float* __restrict__ gamma, const float* __restrict__ bbeta,
    float* __restrict__ bn_s, float* __restrict__ bn_t)
{
    const int h = blockIdx.x * 256 + threadIdx.x;
    float s = 0.f, q = 0.f;
    for (int r = 0; r < NR; ++r) {
        s += psum[(size_t)r * HH + h];
        q += psq [(size_t)r * HH + h];
    }
    const float mu  = s / (float)M1;
    const float var = q / (float)M1 - mu * mu;   // ddof = 0 (population var)
    const float sc  = gamma[h] * rsqrtf(var + BN_EPS);
    bn_s[h] = sc;
    bn_t[h] = bbeta[h] - mu * sc;
}

// ---------------------------------------------------------------------------
// Async global->LDS copy of a 128x16 f32 A tile (2 b128 quads per thread).
// ---------------------------------------------------------------------------
static __device__ __forceinline__ void async_copy_tile(
    const float* __restrict__ st_in, int mBlock, int st_stride, int k,
    uint32_t ldsBase, int q0)
{
#pragma unroll
    for (int qq = 0; qq < 2; ++qq) {
        const int q = q0 + qq;
        const float* gp =
            st_in + (size_t)(mBlock + (q >> 2)) * st_stride + k + (q & 3) * 4;
        const uint32_t lo = ldsBase + (uint32_t)q * 16u;
        asm volatile("global_load_async_to_lds_b128 %0, %1, off"
                     :: "v"(lo), "v"(gp) : "memory");
    }
}

// ---------------------------------------------------------------------------
// One RadLIF timestep, fused:  P = st_prev @ Vz  (M=256,N=1024,K=1024 WMMA)
// + wt/ut/spike update applied straight from the accumulators.
// Spikes live only in `out` ([B,T,H]); step t reads row t-1 (stride T*H).
// A tile double-buffered in LDS via async global->LDS b128 copies so the copy
// of chunk c+1 overlaps the 16 WMMAs of chunk c (serial-scan critical path).
// Block tile 128(M)x64(N), waves 4x2, wave tile 32x32 -> grid (2,16)=32 blocks.
// ---------------------------------------------------------------------------
__global__ __launch_bounds__(256) void step_kernel(
    const float* __restrict__ st_in, int st_stride,
    const float* __restrict__ Vz, const float* __restrict__ Wx,
    const float* __restrict__ alpha_c, const float* __restrict__ beta_c,
    const float* __restrict__ a_c, const float* __restrict__ b_c,
    const float* __restrict__ bn_s, const float* __restrict__ bn_t,
    float* __restrict__ ut, float* __restrict__ wt,
    float* __restrict__ out, int t)
{
    __shared__ __align__(16) float shA[2][128 * 16];   // 2 x 8 KB A tiles

    const int lane   = threadIdx.x & 31;
    const int wid    = threadIdx.x >> 5;
    const int lanelo = lane & 15;
    const int lanehi = lane >> 4;
    const int mBlock = blockIdx.x * 128;
    const int m0w    = mBlock + (wid >> 1) * 32;
    const int n0     = blockIdx.y * 64 + (wid & 1) * 32;

    v8f acc[2][2];
    v8f zero = {0.f, 0.f, 0.f, 0.f, 0.f, 0.f, 0.f, 0.f};
#pragma unroll
    for (int i = 0; i < 2; ++i)
#pragma unroll
        for (int j = 0; j < 2; ++j) acc[i][j] = zero;

    // Low 32 bits of the generic LDS address are the wave-relative LDS offset
    // (aperture occupies ADDR[63:32], ISA 10.2).
    const uint32_t shB0 = (uint32_t)(uintptr_t)&shA[0][0];
    const uint32_t shB1 = (uint32_t)(uintptr_t)&shA[1][0];
    const int q0 = threadIdx.x * 2;   // 512 b128 quads per tile, 2 per thread

    async_copy_tile(st_in, mBlock, st_stride, 0, shB0, q0);

    for (int c = 0; c < HH / 16; ++c) {
        const int k = c * 16;
        if (c + 1 < HH / 16) {
            async_copy_tile(st_in, mBlock, st_stride, k + 16,
                            (c & 1) ? shB0 : shB1, q0);
            // async loads complete in order: <=2 outstanding => chunk c landed
            asm volatile("s_wait_asynccnt 2" ::: "memory");
        } else {
            asm volatile("s_wait_asynccnt 0" ::: "memory");
        }
        __syncthreads();

        const float* buf = &shA[c & 1][0];
#pragma unroll
        for (int kk = 0; kk < 16; kk += 4) {
            const int klane = kk + 2 * lanehi;
            v2f a0 = *(const v2f*)&buf[((wid >> 1) * 32 + lanelo) * 16 + klane];
            v2f a1 = *(const v2f*)&buf[((wid >> 1) * 32 + 16 + lanelo) * 16 + klane];
            const int kg = k + klane;
#pragma unroll
            for (int j = 0; j < 2; ++j) {
                const int n = n0 + j * 16 + lanelo;
                v2f b;
                b.x = Vz[(size_t)kg * HH + n];
                b.y = Vz[(size_t)(kg + 1) * HH + n];
                acc[0][j] = wmma_f32_4(a0, b, acc[0][j]);
                acc[1][j] = wmma_f32_4(a1, b, acc[1][j]);
            }
        }
        __syncthreads();   // all waves done reading before buffer is rewritten
    }

    // Fused RadLIF update (each (b,h) owned by exactly one lane/vgpr slot).
#pragma unroll
    for (int j = 0; j < 2; ++j) {
        const int h  = n0 + j * 16 + lanelo;
        const float al = alpha_c[h], be = beta_c[h];
        const float ac = a_c[h],     bc = b_c[h];
        const float sc = bn_s[h],    sh = bn_t[h];
#pragma unroll
        for (int i = 0; i < 2; ++i) {
            const int mb = m0w + i * 16 + 8 * lanehi;
#pragma unroll
            for (int v = 0; v < 8; ++v) {
                const int bidx = mb + v;
                const size_t sidx = (size_t)bidx * HH + h;
                const float u  = ut[sidx];
                const float w  = wt[sidx];
                const float sp = st_in[(size_t)bidx * st_stride + h];
                const float wx = Wx[((size_t)bidx * TT + t) * HH + h] * sc + sh;
                const float wn = be * w + ac * u + bc * sp;
                const float un = al * (u - sp) + (1.0f - al) * (wx + acc[i][j][v] - wn);
                const float sn = (un > THRESH) ? 1.0f : 0.0f;
                ut[sidx] = un;
                wt[sidx] = wn;
                out[((size_t)bidx * TT + t) * HH + h] = sn;
            }
        }
    }
}

// ---------------------------------------------------------------------------
extern "C" void kernel_launch(void* const* d_in, const int* in_sizes, int n_in,
                              void* d_out, int out_size, void* d_ws, size_t ws_size,
                              hipStream_t stream)
{
    (void)in_sizes; (void)n_in; (void)out_size; (void)ws_size;
    const float* x     = (const float*)d_in[0];
    // d_in[1] = mask (inert)
    const float* W     = (const float*)d_in[2];
    const float* V     = (const float*)d_in[3];
    const float* alpha = (const float*)d_in[4];
    const float* beta  = (const float*)d_in[5];
    const float* aa    = (const float*)d_in[6];
    const float* bb    = (const float*)d_in[7];
    const float* gamma = (const float*)d_in[8];
    const float* bbeta = (const float*)d_in[9];
    const float* ut0   = (const float*)d_in[10];
    const float* wt0   = (const float*)d_in[11];
    const float* st0   = (const float*)d_in[12];
    float* out = (float*)d_out;

    float* ws = (float*)d_ws;
    size_t off = 0;
    float* Wx     = ws + off; off += (size_t)M1 * HH;   // 104.9 MB
    float* Vz     = ws + off; off += (size_t)HH * HH;   // 4 MB
    float* utb    = ws + off; off += (size_t)BB * HH;
    float* wtb    = ws + off; off += (size_t)BB * HH;
    float* alphac = ws + off; off += HH;
    float* betac  = ws + off; off += HH;
    float* ac     = ws + off; off += HH;
    float* bc     = ws + off; off += HH;
    float* bns    = ws + off; off += HH;
    float* bnt    = ws + off; off += HH;
    float* psum   = ws + off; off += (size_t)NR * HH;
    float* psq    = ws + off; off += (size_t)NR * HH;

    prep_kernel<<<(HH * HH) / 256, 256, 0, stream>>>(
        V, alpha, beta, aa, bb, ut0, wt0,
        Vz, utb, wtb, alphac, betac, ac, bc);

    gemm_wx_kernel<<<dim3(M1 / 128, HH / 128), 256, 0, stream>>>(x, W, Wx);

    bn_partial_kernel<<<dim3(HH / 256, NR), 256, 0, stream>>>(Wx, psum, psq);
    bn_final_kernel<<<HH / 256, 256, 0, stream>>>(psum, psq, gamma, bbeta, bns, bnt);

    for (int t = 0; t < TT; ++t) {
        const float* stin = (t == 0) ? st0 : (out + (size_t)(t - 1) * HH);
        const int stride  = (t == 0) ? HH : TT * HH;
        step_kernel<<<dim3(BB / 128, HH / 64), 256, 0, stream>>>(
            stin, stride, Vz, Wx, alphac, betac, ac, bc, bns, bnt,
            utb, wtb, out, t);
    }
}